// Attention_58437325029959
// MI455X (gfx1250) — compile-verified
//
#include <hip/hip_runtime.h>

typedef float v2f __attribute__((ext_vector_type(2)));
typedef float v8f __attribute__((ext_vector_type(8)));

#define B_      8
#define N_      2048
#define D_      128
#define M_      128
#define NIT     50
#define LAMBDA_ 0.1f
#define RHO_    1.0f
#define INV_M   (1.0f/128.0f)
#define PITCH   132
#define GJP     260
#define NT      64          // n-columns per block (4 waves * 16)

__device__ __forceinline__ v8f wmma4(v2f a, v2f b, v8f c) {
  // D = A(16x4,f32) * B(4x16,f32) + C(16x16,f32)  -> v_wmma_f32_16x16x4_f32
  return __builtin_amdgcn_wmma_f32_16x16x4_f32(false, a, false, b, (short)0, c, false, false);
}

// ---------------------------------------------------------------------------
// Kernel 1: per-batch M_inv = inv(2 V' V'^T + rho I), V' = V/128.
// Column-parallel Gauss-Jordan in LDS (G = I + small PSD -> no pivoting).
// ---------------------------------------------------------------------------
__global__ __launch_bounds__(256) void gj_invert(const float* __restrict__ V,
                                                 float* __restrict__ Minv) {
  __shared__ float aug[M_ * GJP];
  __shared__ float colk[M_];
  const int b = blockIdx.x;
  const int tid = threadIdx.x;

  // stage V' into right half
  for (int i = tid; i < M_ * D_; i += 256) {
    int r = i >> 7, c = i & 127;
    aug[r * GJP + 128 + c] = V[(size_t)b * M_ * D_ + i] * INV_M;
  }
  __syncthreads();
  // G into left half
  for (int e = tid; e < M_ * M_; e += 256) {
    int r = e >> 7, c = e & 127;
    float s = 0.0f;
    for (int d = 0; d < D_; ++d)
      s += aug[r * GJP + 128 + d] * aug[c * GJP + 128 + d];
    aug[r * GJP + c] = 2.0f * s + ((r == c) ? RHO_ : 0.0f);
  }
  __syncthreads();
  // identity into right half
  for (int i = tid; i < M_ * M_; i += 256) {
    int r = i >> 7, c = i & 127;
    aug[r * GJP + 128 + c] = (r == c) ? 1.0f : 0.0f;
  }
  __syncthreads();
  // Gauss-Jordan, thread tid owns augmented column tid (0..255)
  const int c = tid;
  for (int k = 0; k < M_; ++k) {
    for (int r = tid; r < M_; r += 256) colk[r] = aug[r * GJP + k];
    __syncthreads();
    float val = aug[k * GJP + c] / colk[k];
    for (int r = 0; r < M_; ++r) {
      float cur = aug[r * GJP + c];
      aug[r * GJP + c] = (r == k) ? val : (cur - colk[r] * val);
    }
    __syncthreads();
  }
  // emit inverse
  for (int i = tid; i < M_ * M_; i += 256) {
    int r = i >> 7, cc = i & 127;
    Minv[(size_t)b * M_ * M_ + i] = aug[r * GJP + 128 + cc];
  }
}

// ---------------------------------------------------------------------------
// Kernel 2: fused P build + 50 ADMM iterations + threshold/normalize + output.
// Block = (batch, 64-column n-tile), 4 waves; wave owns 16 columns x all 128 j.
// Fragment convention (16x16x4 f32 WMMA):
//   A: lane -> row = 16t + (lane&15), vgpr v -> K = 4c + v + 2*(lane>>4)
//   B: lane -> col = lane&15,        vgpr v -> K = 4c + v + 2*(lane>>4)
//   D: lane -> col = lane&15,        vgpr v -> row = 16t + v + 8*(lane>>4)
// ---------------------------------------------------------------------------
__global__ __launch_bounds__(128) void admm_fused(const float* __restrict__ Q,
                                                  const float* __restrict__ V,
                                                  const float* __restrict__ Minv,
                                                  float* __restrict__ Out) {
  __shared__ float ldsA[M_ * PITCH];   // V' then M_inv then V'   [row][col]
  __shared__ float ldsB[NT * PITCH];   // Q-tile then rhs/coeffs  [n_local][k]
  const int tid  = threadIdx.x;
  const int lane = tid & 31;
  const int wave = tid >> 5;
  const int lrow = lane & 15;
  const int hi   = lane >> 4;
  const int bb    = blockIdx.x >> 5;        // batch
  const int nbase = (blockIdx.x & 31) * NT; // global n base of this tile
  const int ncl   = wave * 16 + lrow;       // local column owned by this lane

  const float4* Q4 = (const float4*)(Q + ((size_t)bb * N_ + nbase) * D_);
  const float4* V4 = (const float4*)(V + (size_t)bb * M_ * D_);

  // ---- stage Q tile [64][128] -> ldsB, V' [128][128] -> ldsA (coalesced) ----
  for (int i = tid; i < NT * (D_ / 4); i += 128) {
    int r = i >> 5, c4 = i & 31;
    float4 q = Q4[i];
    float* d = &ldsB[r * PITCH + c4 * 4];
    d[0] = q.x; d[1] = q.y; d[2] = q.z; d[3] = q.w;
  }
  for (int i = tid; i < M_ * (D_ / 4); i += 128) {
    int r = i >> 5, c4 = i & 31;
    float4 vv = V4[i];
    float* d = &ldsA[r * PITCH + c4 * 4];
    d[0] = vv.x * INV_M; d[1] = vv.y * INV_M; d[2] = vv.z * INV_M; d[3] = vv.w * INV_M;
  }
  __syncthreads();

  // ---- P_t[j][n] = (V' Q^T)[j][n] fragments (to be scaled) ----
  v8f P[8];
  #pragma unroll
  for (int t = 0; t < 8; ++t)
    #pragma unroll
    for (int e = 0; e < 8; ++e) P[t][e] = 0.0f;
  for (int c = 0; c < 32; ++c) {
    v2f bf = *(const v2f*)&ldsB[ncl * PITCH + 4 * c + 2 * hi];       // Q[n][d]
    #pragma unroll
    for (int t = 0; t < 8; ++t) {
      v2f af = *(const v2f*)&ldsA[(16 * t + lrow) * PITCH + 4 * c + 2 * hi]; // V'[j][d]
      P[t] = wmma4(af, bf, P[t]);
    }
  }

  // finalize P, u = 0, initial rhs = -P written to wave-private ldsB rows
  v8f u[8];
  #pragma unroll
  for (int t = 0; t < 8; ++t)
    #pragma unroll
    for (int e = 0; e < 8; ++e) {
      float p = -2.0f * P[t][e] + (LAMBDA_ / (float)M_);
      P[t][e] = p;
      u[t][e] = 0.0f;
      ldsB[ncl * PITCH + 16 * t + 8 * hi + e] = -p;
    }
  __syncthreads();                       // done reading V' from ldsA

  // ---- M_inv -> ldsA ----
  const float4* MI4 = (const float4*)(Minv + (size_t)bb * M_ * M_);
  for (int i = tid; i < M_ * (M_ / 4); i += 128) {
    int r = i >> 5, c4 = i & 31;
    float4 m = MI4[i];
    float* d = &ldsA[r * PITCH + c4 * 4];
    d[0] = m.x; d[1] = m.y; d[2] = m.z; d[3] = m.w;
  }
  __syncthreads();

  // ---- 50 ADMM iterations, barrier-free (wave-private rhs rows in ldsB) ----
  v8f x[8];
  for (int it = 0; it < NIT; ++it) {
    #pragma unroll
    for (int t = 0; t < 8; ++t)
      #pragma unroll
      for (int e = 0; e < 8; ++e) x[t][e] = 0.0f;
    for (int c = 0; c < 32; ++c) {
      v2f bf = *(const v2f*)&ldsB[ncl * PITCH + 4 * c + 2 * hi];     // rhs[j][n]
      #pragma unroll
      for (int t = 0; t < 8; ++t) {
        v2f af = *(const v2f*)&ldsA[(16 * t + lrow) * PITCH + 4 * c + 2 * hi]; // Minv[i][j]
        x[t] = wmma4(af, bf, x[t]);
      }
    }
    if (it < NIT - 1) {
      #pragma unroll
      for (int t = 0; t < 8; ++t)
        #pragma unroll
        for (int e = 0; e < 8; ++e) {
          float xn = x[t][e] + u[t][e];
          float z  = fminf(fmaxf(xn, 0.0f), 1.0f);
          float un = xn - z;
          u[t][e]  = un;
          ldsB[ncl * PITCH + 16 * t + 8 * hi + e] = RHO_ * (z - un) - P[t][e];
        }
    } else {
      #pragma unroll
      for (int t = 0; t < 8; ++t)
        #pragma unroll
        for (int e = 0; e < 8; ++e) {
          float xn = x[t][e] + u[t][e];
          x[t][e]  = fminf(fmaxf(xn, 0.0f), 1.0f);   // final z
        }
    }
  }

  // ---- threshold + column-normalize -> coeff fragments ----
  float cnt = 0.0f;
  #pragma unroll
  for (int t = 0; t < 8; ++t)
    #pragma unroll
    for (int e = 0; e < 8; ++e) {
      float bit = (x[t][e] > 0.5f) ? 1.0f : 0.0f;
      x[t][e] = bit;
      cnt += bit;
    }
  cnt += __shfl_xor(cnt, 16, 32);        // lanes l and l+16 share column n
  float coef = 1.0f / (cnt + 1e-10f);

  __syncthreads();                       // all waves done reading Minv
  // reload V' -> ldsA, coeffs -> wave-private ldsB rows
  for (int i = tid; i < M_ * (D_ / 4); i += 128) {
    int r = i >> 5, c4 = i & 31;
    float4 vv = V4[i];
    float* d = &ldsA[r * PITCH + c4 * 4];
    d[0] = vv.x * INV_M; d[1] = vv.y * INV_M; d[2] = vv.z * INV_M; d[3] = vv.w * INV_M;
  }
  #pragma unroll
  for (int t = 0; t < 8; ++t)
    #pragma unroll
    for (int e = 0; e < 8; ++e)
      ldsB[ncl * PITCH + 16 * t + 8 * hi + e] = x[t][e] * coef;
  __syncthreads();

  // ---- Out^T[dd][n] = sum_j V'[j][dd] * C[j][n] ----
  v8f o[8];
  #pragma unroll
  for (int t = 0; t < 8; ++t)
    #pragma unroll
    for (int e = 0; e < 8; ++e) o[t][e] = 0.0f;
  for (int c = 0; c < 32; ++c) {
    v2f bf = *(const v2f*)&ldsB[ncl * PITCH + 4 * c + 2 * hi];       // C[j][n]
    #pragma unroll
    for (int t = 0; t < 8; ++t) {
      v2f af;                                                        // V'^T[dd][j]
      af.x = ldsA[(4 * c + 0 + 2 * hi) * PITCH + 16 * t + lrow];
      af.y = ldsA[(4 * c + 1 + 2 * hi) * PITCH + 16 * t + lrow];
      o[t] = wmma4(af, bf, o[t]);
    }
  }
  // store: lane's column n = nbase+ncl, rows dd = 16t + 8hi + (0..7) contiguous
  float* outp = Out + ((size_t)bb * N_ + nbase + ncl) * D_;
  #pragma unroll
  for (int t = 0; t < 8; ++t) {
    float4 s0 = make_float4(o[t][0], o[t][1], o[t][2], o[t][3]);
    float4 s1 = make_float4(o[t][4], o[t][5], o[t][6], o[t][7]);
    *(float4*)(outp + 16 * t + 8 * hi)     = s0;
    *(float4*)(outp + 16 * t + 8 * hi + 4) = s1;
  }
}

extern "C" void kernel_launch(void* const* d_in, const int* in_sizes, int n_in,
                              void* d_out, int out_size, void* d_ws, size_t ws_size,
                              hipStream_t stream) {
  const float* Q = (const float*)d_in[0];   // [8,2048,128] f32
  const float* V = (const float*)d_in[1];   // [8,128,128]  f32
  float* Out  = (float*)d_out;              // [8,2048,128] f32
  float* Minv = (float*)d_ws;               // 8*128*128 f32 = 512 KB scratch

  gj_invert<<<dim3(B_), dim3(256), 0, stream>>>(V, Minv);
  admm_fused<<<dim3(B_ * (N_ / NT)), dim3(128), 0, stream>>>(Q, V, Minv, Out);
}